// SKA_44830868635847
// MI455X (gfx1250) — compile-verified
//
#include <hip/hip_runtime.h>
#include <stdint.h>

// SKA: per-pixel dynamic 3x3 conv, weights shared across 8 group channels.
// B=8, C=64, H=W=128, C_w=8, groups=8, ks=3, pad=1.
//
// Memory-bound (AI ~1.4 FLOP/B -> HBM roofline ~4.5us at 23.3 TB/s).
// Strategy: async global->LDS staging of the x tile (CDNA5
// GLOBAL_LOAD_ASYNC_TO_LDS_B128, ASYNCcnt), per-pixel weights kept in VGPRs
// and reused across the 8 group channels, float4 (b128) transactions
// throughout, wave32-friendly 256-thread blocks.

#define AS1P __attribute__((address_space(1)))
#define AS3P __attribute__((address_space(3)))

typedef int v4i __attribute__((ext_vector_type(4)));

__device__ __forceinline__ void async_ld_f4(const float* gsrc, float* ldst) {
#if __has_builtin(__builtin_amdgcn_global_load_async_to_lds_b128)
  // Params: (global v4i*, lds v4i*, imm offset, imm cache-policy).
  __builtin_amdgcn_global_load_async_to_lds_b128(
      (AS1P v4i*)(uintptr_t)gsrc, (AS3P v4i*)(uintptr_t)ldst, 0, 0);
#else
  unsigned l = (unsigned)(uintptr_t)ldst;  // low 32 bits of flat LDS addr = LDS offset
  asm volatile("global_load_async_to_lds_b128 %0, %1, off"
               :: "v"(l), "v"(gsrc) : "memory");
#endif
}

__device__ __forceinline__ void wait_async0() {
#if __has_builtin(__builtin_amdgcn_s_wait_asynccnt)
  __builtin_amdgcn_s_wait_asynccnt(0);
#else
  asm volatile("s_wait_asynccnt 0" ::: "memory");
#endif
}

constexpr int kB = 8, kC = 64, kH = 128, kW = 128, kCW = 8, kG = 8;
constexpr int kTH = 8;             // output rows per block
constexpr int kLR = kTH + 2;       // staged rows (with vertical halo)

__global__ __launch_bounds__(256) void ska_kernel(
    const float* __restrict__ x, const float* __restrict__ wt,
    float* __restrict__ out) {
  __shared__ float tile[kG * kLR * kW];  // 8ch * 10 rows * 128 = 40 KB

  const int tid = threadIdx.x;
  const int h0 = blockIdx.x * kTH;
  const int cw = blockIdx.y;
  const int b  = blockIdx.z;

  const int rslot = tid >> 5;        // 0..7 (row slot; one wave per row)
  const int c4    = (tid & 31) << 2; // float column, 16B aligned

  // Stage one (ch,row) of the x tile: image row h0-1+row, full 128-wide.
  auto stage = [&](int ch, int row) {
    int gr = h0 - 1 + row;  // global image row
    float* ldst = &tile[(ch * kLR + row) * kW + c4];
    if ((unsigned)gr < (unsigned)kH) {
      const float* gsrc =
          x + (((b * kC + ch * kCW + cw) * kH + gr) * kW + c4);
      async_ld_f4(gsrc, ldst);
    } else {
      // zero padding rows at the image top/bottom
      *reinterpret_cast<float4*>(ldst) = float4{0.f, 0.f, 0.f, 0.f};
    }
  };

  // ---- Stage x tile: 8 channels x 10 rows x 128 cols via async DMA to LDS.
  // Iterations 0..7: rows 0..7 of channel `it` (one row per wave row-slot).
#pragma unroll
  for (int it = 0; it < kG; ++it) stage(it, rslot);
  // Tail: 16 halo rows (rows 8,9 of each channel), shift/mask mapping only.
#pragma unroll
  for (int it = 0; it < 2; ++it) {
    int idx = it * 8 + rslot;        // 0..15
    stage(idx >> 1, 8 + (idx & 1));
  }

  // ---- Per-pixel weights: 9 taps x float4 into VGPRs (overlaps async DMA).
  const int r  = rslot;
  const int hg = h0 + r;
  const float* wb =
      wt + (((size_t)(b * kCW + cw) * 9) * kH + hg) * kW + c4;
  float4 wk[9];
#pragma unroll
  for (int k = 0; k < 9; ++k)
    wk[k] = *reinterpret_cast<const float4*>(wb + (size_t)k * kH * kW);

  wait_async0();      // all of this wave's async DMA landed in LDS
  __syncthreads();    // all waves' tiles visible

  // ---- Compute: for each group channel, 3x3 tap accumulation on 4 pixels.
#pragma unroll
  for (int g = 0; g < kG; ++g) {
    float4 acc = {0.f, 0.f, 0.f, 0.f};
#pragma unroll
    for (int i = 0; i < 3; ++i) {
      const float* rp = &tile[(g * kLR + r + i) * kW];  // row h0+r-1+i
      float4 xm = *reinterpret_cast<const float4*>(rp + c4);
      float xl = (c4 == 0)      ? 0.f : rp[c4 - 1];  // image left pad
      float xr = (c4 == kW - 4) ? 0.f : rp[c4 + 4];  // image right pad
      float4 w0 = wk[3 * i + 0], w1 = wk[3 * i + 1], w2 = wk[3 * i + 2];
      acc.x = fmaf(xl,   w0.x, fmaf(xm.x, w1.x, fmaf(xm.y, w2.x, acc.x)));
      acc.y = fmaf(xm.x, w0.y, fmaf(xm.y, w1.y, fmaf(xm.z, w2.y, acc.y)));
      acc.z = fmaf(xm.y, w0.z, fmaf(xm.z, w1.z, fmaf(xm.w, w2.z, acc.z)));
      acc.w = fmaf(xm.z, w0.w, fmaf(xm.w, w1.w, fmaf(xr,   w2.w, acc.w)));
    }
    float* op = out + (((b * kC + g * kCW + cw) * kH + hg) * kW + c4);
    *reinterpret_cast<float4*>(op) = acc;
  }
}

extern "C" void kernel_launch(void* const* d_in, const int* in_sizes, int n_in,
                              void* d_out, int out_size, void* d_ws,
                              size_t ws_size, hipStream_t stream) {
  const float* x  = (const float*)d_in[0];  // (8,64,128,128)
  const float* wt = (const float*)d_in[1];  // (8,8,9,128,128)
  float* out = (float*)d_out;               // (8,64,128,128)
  (void)in_sizes; (void)n_in; (void)out_size; (void)d_ws; (void)ws_size;

  dim3 grid(kH / kTH, kCW, kB);  // (16, 8, 8)
  ska_kernel<<<grid, 256, 0, stream>>>(x, wt, out);
}